// linear_update_5411658793255
// MI455X (gfx1250) — compile-verified
//
#include <hip/hip_runtime.h>
#include <cstdint>

// Problem constants from the reference.
#define B_DIM  4096
#define T_DIM  256
#define NC_DIM 16
#define NR_DIM 32
#define D_DIM  64   // 2*NC + NR

typedef float f32x4 __attribute__((ext_vector_type(4)));

union Half32 {
    f32x4 v[8];
    float f[32];
};

// One block per batch b (grid.x = 4096), one thread per time step t (block = 256).
// Each thread produces one 64-float output row:
//   out[b,t,0:16]   = y0*re + y1*im          (complex rotation, per pair c)
//   out[b,t,16:32]  = y1*re - y0*im
//   out[b,t,32:64]  = y[b,0,32+r] * exp(Km[b,0,32+r] * t)
__global__ __launch_bounds__(T_DIM, 1) void rot_decay_kernel(
    const float* __restrict__ y,
    const float* __restrict__ Km,
    float* __restrict__ out)
{
    // smem[0:32)  = y[b, 0, 32:64]   (R-part initial values)
    // smem[32:64) = Km[b, 0, 32:64]  (log decay rates)
    __shared__ float smem[2 * NR_DIM];

    const int b = blockIdx.x;
    const int t = threadIdx.x;
    const size_t batchBase = (size_t)b * T_DIM * D_DIM;

    // --- CDNA5 async copy: pull the 256B of per-batch constants into LDS ---
    // Lanes 0..7  copy y [b,0,32:64] (8 x 16B)  -> smem[0:32)
    // Lanes 8..15 copy Km[b,0,32:64] (8 x 16B)  -> smem[32:64)
    if (threadIdx.x < 16u) {
        const unsigned lane = threadIdx.x;
        const float* gbase = (lane & 8u) ? (Km + batchBase + 2 * NC_DIM)
                                         : (y  + batchBase + 2 * NC_DIM);
        const float* gaddr = gbase + 4u * (lane & 7u);
        // Low 32 bits of a generic pointer to LDS == LDS byte offset.
        unsigned ldsoff = (unsigned)(uintptr_t)(&smem[4u * lane]);
        asm volatile("global_load_async_to_lds_b128 %0, %1, off"
                     :: "v"(ldsoff), "v"(gaddr) : "memory");
        asm volatile("s_wait_asynccnt 0" ::: "memory");
    }
    __syncthreads();

    // --- Stream in this thread's row halves (read-once -> non-temporal) ---
    const size_t row = batchBase + (size_t)t * D_DIM;
    const f32x4* yp = (const f32x4*)(y  + row);
    const f32x4* kp = (const f32x4*)(Km + row);

    Half32 yv, kv;
#pragma unroll
    for (int i = 0; i < 8; ++i) {
        yv.v[i] = __builtin_nontemporal_load(yp + i);
        kv.v[i] = __builtin_nontemporal_load(kp + i);
    }

    // --- C part: per-pair complex rotation ---
    Half32 oc;
#pragma unroll
    for (int c = 0; c < NC_DIM; ++c) {
        const float ya = yv.f[2 * c];
        const float yb = yv.f[2 * c + 1];
        const float s  = __expf(kv.f[c]);          // exp(logs)
        float sn, cs;
        __sincosf(kv.f[NC_DIM + c], &sn, &cs);     // sin/cos(theta)
        const float re = s * cs;
        const float im = s * sn;
        oc.f[c]          = ya * re + yb * im;
        oc.f[NC_DIM + c] = yb * re - ya * im;
    }

    // --- R part: per-channel exponential decay from LDS-broadcast constants ---
    Half32 orr;
    const float tf = (float)t;
#pragma unroll
    for (int r = 0; r < NR_DIM; ++r) {
        orr.f[r] = smem[r] * __expf(smem[NR_DIM + r] * tf);
    }

    // --- Write-once output -> non-temporal b128 stores ---
    f32x4* op = (f32x4*)(out + row);
#pragma unroll
    for (int i = 0; i < 8; ++i) {
        __builtin_nontemporal_store(oc.v[i], op + i);
    }
#pragma unroll
    for (int i = 0; i < 8; ++i) {
        __builtin_nontemporal_store(orr.v[i], op + 8 + i);
    }
}

extern "C" void kernel_launch(void* const* d_in, const int* in_sizes, int n_in,
                              void* d_out, int out_size, void* d_ws, size_t ws_size,
                              hipStream_t stream) {
    (void)in_sizes; (void)n_in; (void)out_size; (void)d_ws; (void)ws_size;
    const float* y  = (const float*)d_in[0];
    const float* Km = (const float*)d_in[1];
    float* out = (float*)d_out;
    rot_decay_kernel<<<dim3(B_DIM), dim3(T_DIM), 0, stream>>>(y, Km, out);
}